// ContactLoss_44057774522464
// MI455X (gfx1250) — compile-verified
//
#include <hip/hip_runtime.h>
#include <hip/hip_bf16.h>
#include <cmath>

typedef float v2f __attribute__((ext_vector_type(2)));
typedef float v8f __attribute__((ext_vector_type(8)));

#define RAYX 0.4395064455f
#define RAYY 0.617598629942f
#define RAYZ 0.652231566745f
#define TOLF 1e-7f
#define BIGF 1e10f
#define HUGEF 3.0e38f

// ---------------------------------------------------------------------------
// Per-triangle precompute: v0, e1, e2, pvec=cross(dir,e2), invdet, valid flag.
// 16 floats per triangle.
// ---------------------------------------------------------------------------
__global__ void tri_pre_kernel(const float* __restrict__ verts, int Nv,
                               const int* __restrict__ faces, int F,
                               const int* __restrict__ splits,  // null => all valid
                               float* __restrict__ out) {
  int b = blockIdx.y;
  int t = blockIdx.x * blockDim.x + threadIdx.x;
  if (t >= F) return;
  const int* f = faces + ((size_t)b * F + t) * 3;
  const float* vb = verts + (size_t)b * Nv * 3;
  int i0 = f[0], i1 = f[1], i2 = f[2];
  float v0x = vb[i0*3+0], v0y = vb[i0*3+1], v0z = vb[i0*3+2];
  float v1x = vb[i1*3+0], v1y = vb[i1*3+1], v1z = vb[i1*3+2];
  float v2x = vb[i2*3+0], v2y = vb[i2*3+1], v2z = vb[i2*3+2];
  float e1x = v1x - v0x, e1y = v1y - v0y, e1z = v1z - v0z;
  float e2x = v2x - v0x, e2y = v2y - v0y, e2z = v2z - v0z;
  float pvx = RAYY * e2z - RAYZ * e2y;
  float pvy = RAYZ * e2x - RAYX * e2z;
  float pvz = RAYX * e2y - RAYY * e2x;
  float det = e1x*pvx + e1y*pvy + e1z*pvz;
  bool parallel = fabsf(det) < TOLF;
  float invdet = 1.0f / (det + 0.1f * TOLF);
  bool valid = (!parallel) && (splits ? (t < splits[b]) : true);
  float* o = out + ((size_t)b * F + t) * 16;
  o[0]=v0x;  o[1]=v0y;  o[2]=v0z;
  o[3]=e1x;  o[4]=e1y;  o[5]=e1z;
  o[6]=e2x;  o[7]=e2y;  o[8]=e2z;
  o[9]=pvx;  o[10]=pvy; o[11]=pvz;
  o[12]=invdet; o[13]= valid ? 1.0f : 0.0f; o[14]=0.f; o[15]=0.f;
}

// ---------------------------------------------------------------------------
// Moller-Trumbore parity ray-cast: each thread = one point, triangles staged
// through LDS in tiles of 64 (4 KB).
// ---------------------------------------------------------------------------
#define TRI_TILE 64
__global__ void raycast_kernel(const float* __restrict__ points, int P,
                               const float* __restrict__ tris, int T,
                               int* __restrict__ exterior) {
  int b = blockIdx.y;
  int p = blockIdx.x * blockDim.x + threadIdx.x;
  const float* pb = points + (size_t)b * P * 3;
  const float* tb = tris + (size_t)b * T * 16;
  int pp = p < P ? p : P - 1;
  float px = pb[pp*3+0], py = pb[pp*3+1], pz = pb[pp*3+2];
  __shared__ float tile[TRI_TILE * 16];
  int hits = 0;
  for (int t0 = 0; t0 < T; t0 += TRI_TILE) {
    int cnt = min(TRI_TILE, T - t0);
    __syncthreads();
    for (int i = threadIdx.x; i < cnt * 16; i += blockDim.x)
      tile[i] = tb[(size_t)t0 * 16 + i];
    __syncthreads();
    for (int t = 0; t < cnt; ++t) {
      const float* tr = &tile[t * 16];
      float tvx = px - tr[0], tvy = py - tr[1], tvz = pz - tr[2];
      float inv = tr[12];
      float u  = (tvx*tr[9] + tvy*tr[10] + tvz*tr[11]) * inv;
      float qx = tvy*tr[5] - tvz*tr[4];
      float qy = tvz*tr[3] - tvx*tr[5];
      float qz = tvx*tr[4] - tvy*tr[3];
      float v  = (qx*RAYX + qy*RAYY + qz*RAYZ) * inv;
      float tt = (tr[6]*qx + tr[7]*qy + tr[8]*qz) * inv;
      bool hit = (u > 0.f) && (u < 1.f) && (v > 0.f) && ((u + v) < 1.f)
                 && (tt > TOLF) && (tr[13] != 0.f);
      hits += hit ? 1 : 0;
    }
  }
  if (p < P) exterior[(size_t)b * P + p] = ((hits & 1) == 0) ? 1 : 0;
}

// ---------------------------------------------------------------------------
// minho: for each hand row, argmin over obj columns of rx+ry-2*zz (invalid obj
// columns masked to BIG). One wave per 16 hand rows; zz tiles come from
// V_WMMA_F32_16X16X4_F32 (K=3 padded to 4).
// C/D layout: VGPR g, lanes<16 -> row g, lanes>=16 -> row g+8; col = lane&15.
// ---------------------------------------------------------------------------
__global__ void minho_kernel(const float* __restrict__ hand, int Nh,
                             const float* __restrict__ obj, int No,
                             const int* __restrict__ splits,
                             int* __restrict__ minho_idx) {
  int b = blockIdx.y;
  int hbase = blockIdx.x * 16;
  int lane = threadIdx.x;
  int l16 = lane & 15, half = lane >> 4;
  const float* hb = hand + (size_t)b * Nh * 3;
  const float* ob = obj + (size_t)b * No * 3;

  int m = hbase + l16;
  int mm = m < Nh ? m : Nh - 1;
  float hx = hb[mm*3+0], hy = hb[mm*3+1], hz = hb[mm*3+2];
  v2f A;                               // A 16x4: lanes<16 K0,K1 ; lanes>=16 K2,K3
  A.x = half ? hz : hx;
  A.y = half ? 0.0f : hy;

  __shared__ float rxs[16];
  if (half == 0) rxs[l16] = hx*hx + hy*hy + hz*hz;
  __syncthreads();
  float rxg[8];
#pragma unroll
  for (int g = 0; g < 8; ++g) rxg[g] = rxs[g + 8*half];

  int split = splits[b];
  float best[8]; int bidx[8];
#pragma unroll
  for (int g = 0; g < 8; ++g) { best[g] = HUGEF; bidx[g] = 0; }

  int ntiles = (No + 15) / 16;
  for (int ot = 0; ot < ntiles; ++ot) {
    int n = ot * 16 + l16;
    int nn = n < No ? n : No - 1;
    float ox = ob[nn*3+0], oy = ob[nn*3+1], oz = ob[nn*3+2];
    v2f Bm;                            // B 4x16: VGPR j, half h -> K = j + 2h
    Bm.x = half ? oz : ox;
    Bm.y = half ? 0.0f : oy;
    v8f C = {};
    C = __builtin_amdgcn_wmma_f32_16x16x4_f32(false, A, false, Bm,
                                              (short)0, C, false, false);
    float ry = ox*ox + oy*oy + oz*oz;
    bool exists = n < No;
    bool valid = exists && (n < split);
#pragma unroll
    for (int g = 0; g < 8; ++g) {
      float d = exists ? (valid ? (rxg[g] + ry - 2.0f * C[g]) : BIGF) : HUGEF;
      if (d < best[g]) { best[g] = d; bidx[g] = n; }
    }
  }
  // reduce across the 16 lanes of each half (xor masks < 16 stay in-half)
#pragma unroll
  for (int off = 1; off <= 8; off <<= 1) {
#pragma unroll
    for (int g = 0; g < 8; ++g) {
      float ov = __shfl_xor(best[g], off, 32);
      int   oi = __shfl_xor(bidx[g], off, 32);
      if (ov < best[g] || (ov == best[g] && oi < bidx[g])) { best[g]=ov; bidx[g]=oi; }
    }
  }
  if (l16 == 0) {
#pragma unroll
    for (int g = 0; g < 8; ++g) {
      int row = hbase + g + 8*half;
      if (row < Nh) minho_idx[(size_t)b * Nh + row] = bidx[g];
    }
  }
}

// ---------------------------------------------------------------------------
// minoh: for each obj column, argmin over hand rows. One wave per 16 obj cols;
// loops over hand tiles, same WMMA; column-min = min over 8 accum VGPRs then
// xor-16 across lane halves.
// ---------------------------------------------------------------------------
__global__ void minoh_kernel(const float* __restrict__ hand, int Nh,
                             const float* __restrict__ obj, int No,
                             int* __restrict__ minoh_idx) {
  int b = blockIdx.y;
  int obase = blockIdx.x * 16;
  int lane = threadIdx.x;
  int l16 = lane & 15, half = lane >> 4;
  const float* hb = hand + (size_t)b * Nh * 3;
  const float* ob = obj + (size_t)b * No * 3;

  int n = obase + l16;
  int nn = n < No ? n : No - 1;
  float ox = ob[nn*3+0], oy = ob[nn*3+1], oz = ob[nn*3+2];
  float ry = ox*ox + oy*oy + oz*oz;
  v2f Bm;
  Bm.x = half ? oz : ox;
  Bm.y = half ? 0.0f : oy;

  __shared__ float rxs[16];
  float best = HUGEF; int bidx = 0;

  int htiles = (Nh + 15) / 16;
  for (int ht = 0; ht < htiles; ++ht) {
    int ml = ht * 16 + l16;
    int mm = ml < Nh ? ml : Nh - 1;
    float hx = hb[mm*3+0], hy = hb[mm*3+1], hz = hb[mm*3+2];
    v2f A;
    A.x = half ? hz : hx;
    A.y = half ? 0.0f : hy;
    __syncthreads();                       // WAR on rxs from previous iter
    if (half == 0) rxs[l16] = hx*hx + hy*hy + hz*hz;
    __syncthreads();                       // RAW
    v8f C = {};
    C = __builtin_amdgcn_wmma_f32_16x16x4_f32(false, A, false, Bm,
                                              (short)0, C, false, false);
#pragma unroll
    for (int g = 0; g < 8; ++g) {
      int mrow = ht * 16 + g + 8*half;
      float d = (mrow < Nh) ? (rxs[g + 8*half] + ry - 2.0f * C[g]) : HUGEF;
      if (d < best) { best = d; bidx = mrow; }
    }
  }
  float ov = __shfl_xor(best, 16, 32);
  int   oi = __shfl_xor(bidx, 16, 32);
  if (ov < best || (ov == best && oi < bidx)) { best = ov; bidx = oi; }
  if (half == 0 && n < No) minoh_idx[(size_t)b * No + n] = bidx;
}

// ---------------------------------------------------------------------------
// Per-batch masked sums: {sum_missed, cnt_missed, sum_pen_h, cnt_pen_h,
//                         sum_pen_o, cnt_pen_o}. One block per batch.
// ---------------------------------------------------------------------------
__global__ void losses_kernel(const float* __restrict__ hand, int Nh,
                              const float* __restrict__ obj, int No,
                              const int* __restrict__ splits,
                              const int* __restrict__ ext_hand,
                              const int* __restrict__ ext_obj,
                              const int* __restrict__ minho_idx,
                              const int* __restrict__ minoh_idx,
                              float* __restrict__ partials) {
  int b = blockIdx.x;
  int tid = threadIdx.x;
  const float* hb = hand + (size_t)b * Nh * 3;
  const float* ob = obj + (size_t)b * No * 3;
  int split = splits[b];
  float acc[6] = {0.f, 0.f, 0.f, 0.f, 0.f, 0.f};

  for (int i = tid; i < Nh; i += blockDim.x) {
    int idx = minho_idx[(size_t)b * Nh + i];
    float dx = ob[idx*3+0] - hb[i*3+0];
    float dy = ob[idx*3+1] - hb[i*3+1];
    float dz = ob[idx*3+2] - hb[i*3+2];
    float d  = sqrtf(dx*dx + dy*dy + dz*dz);
    float val = 25.0f * tanhf(d * (1.0f / 25.0f));  // CONTACT==COLLISION==25
    if (ext_hand[(size_t)b * Nh + i]) { acc[0] += val; acc[1] += 1.f; }
    else                              { acc[2] += val; acc[3] += 1.f; }
  }
  for (int j = tid; j < No; j += blockDim.x) {
    bool valid = j < split;
    if (valid && !ext_obj[(size_t)b * No + j]) {
      int idx = minoh_idx[(size_t)b * No + j];
      float dx = hb[idx*3+0] - ob[j*3+0];
      float dy = hb[idx*3+1] - ob[j*3+1];
      float dz = hb[idx*3+2] - ob[j*3+2];
      float d  = sqrtf(dx*dx + dy*dy + dz*dz);
      acc[4] += 25.0f * tanhf(d * (1.0f / 25.0f));
      acc[5] += 1.f;
    }
  }
  __shared__ float buf[256];
  for (int k = 0; k < 6; ++k) {
    __syncthreads();
    buf[tid] = acc[k];
    __syncthreads();
    for (int s = blockDim.x >> 1; s > 0; s >>= 1) {
      if (tid < s) buf[tid] += buf[tid + s];
      __syncthreads();
    }
    if (tid == 0) partials[b * 6 + k] = buf[0];
  }
}

// ---------------------------------------------------------------------------
// Finalize: 18 outputs = [missed_loss, penetr_loss, missed_losses[B], penetr_losses[B]]
// ---------------------------------------------------------------------------
__global__ void finalize_kernel(const float* __restrict__ partials,
                                float* __restrict__ out, int B) {
  if (threadIdx.x != 0 || blockIdx.x != 0) return;
  float tsm=0, tcm=0, tsh=0, tch=0, tso=0, tco=0;
  for (int b = 0; b < B; ++b) {
    float sm = partials[b*6+0], cm = partials[b*6+1];
    float sh = partials[b*6+2], ch = partials[b*6+3];
    float so = partials[b*6+4], co = partials[b*6+5];
    out[2 + b] = (cm > 0.f) ? sm / fmaxf(cm, 1.0f) : 0.0f;
    float ph = (ch > 0.f) ? sh / fmaxf(ch, 1.0f) : 0.0f;
    float po = (co > 0.f) ? so / fmaxf(co, 1.0f) : 0.0f;
    out[2 + B + b] = ph + po;
    tsm += sm; tcm += cm; tsh += sh; tch += ch; tso += so; tco += co;
  }
  out[0] = (tcm > 0.f) ? tsm / fmaxf(tcm, 1.0f) : 0.0f;
  float gh = (tch > 0.f) ? tsh / fmaxf(tch, 1.0f) : 0.0f;
  float go = (tco > 0.f) ? tso / fmaxf(tco, 1.0f) : 0.0f;
  out[1] = gh + go;
}

// ---------------------------------------------------------------------------
extern "C" void kernel_launch(void* const* d_in, const int* in_sizes, int n_in,
                              void* d_out, int out_size, void* d_ws, size_t ws_size,
                              hipStream_t stream) {
  const float* hand_verts = (const float*)d_in[0];
  const int*   hand_faces = (const int*)d_in[1];
  const float* obj_verts  = (const float*)d_in[2];
  const int*   obj_faces  = (const int*)d_in[3];
  const int*   vert_splits = (const int*)d_in[4];
  const int*   face_splits = (const int*)d_in[5];

  int B  = in_sizes[4];
  int Nh = in_sizes[0] / (3 * B);
  int Fh = in_sizes[1] / (3 * B);
  int No = in_sizes[2] / (3 * B);
  int Fo = in_sizes[3] / (3 * B);

  float* ws = (float*)d_ws;
  size_t off = 0;
  float* hand_tri = ws + off; off += (size_t)B * Fh * 16;
  float* obj_tri  = ws + off; off += (size_t)B * Fo * 16;
  int* ext_obj  = (int*)(ws + off); off += (size_t)B * No;
  int* ext_hand = (int*)(ws + off); off += (size_t)B * Nh;
  int* minho    = (int*)(ws + off); off += (size_t)B * Nh;
  int* minoh    = (int*)(ws + off); off += (size_t)B * No;
  float* partials = ws + off; off += (size_t)B * 6;

  // 1. triangle precompute (hand tris always valid; obj tris masked by split)
  tri_pre_kernel<<<dim3((Fh + 255) / 256, B), 256, 0, stream>>>(
      hand_verts, Nh, hand_faces, Fh, nullptr, hand_tri);
  tri_pre_kernel<<<dim3((Fo + 255) / 256, B), 256, 0, stream>>>(
      obj_verts, No, obj_faces, Fo, face_splits, obj_tri);

  // 2. parity ray-casts
  raycast_kernel<<<dim3((No + 255) / 256, B), 256, 0, stream>>>(
      obj_verts, No, hand_tri, Fh, ext_obj);
  raycast_kernel<<<dim3((Nh + 255) / 256, B), 256, 0, stream>>>(
      hand_verts, Nh, obj_tri, Fo, ext_hand);

  // 3. WMMA distance-matrix argmins
  minho_kernel<<<dim3((Nh + 15) / 16, B), 32, 0, stream>>>(
      hand_verts, Nh, obj_verts, No, vert_splits, minho);
  minoh_kernel<<<dim3((No + 15) / 16, B), 32, 0, stream>>>(
      hand_verts, Nh, obj_verts, No, minoh);

  // 4. masked means
  losses_kernel<<<dim3(B), 256, 0, stream>>>(
      hand_verts, Nh, obj_verts, No, vert_splits,
      ext_hand, ext_obj, minho, minoh, partials);
  finalize_kernel<<<dim3(1), 32, 0, stream>>>(partials, (float*)d_out, B);
}